// SelFeatTgt_71768903516642
// MI455X (gfx1250) — compile-verified
//
#include <hip/hip_runtime.h>
#include <hip/hip_bf16.h>

typedef __attribute__((ext_vector_type(2))) float v2f;
typedef __attribute__((ext_vector_type(8))) float v8f;

#define M_PTS   4096               // K*C = 128*32 query points
#define N_PTS   32768              // target points
#define F_DIM   256                // feature dim
#define NSAMPLE 32                 // neighbors
#define NCHUNK  8                  // N split for parallel partial top-k
#define CHUNK_N (N_PTS / NCHUNK)   // 4096
#define NTILES  (CHUNK_N / 16)     // 256 WMMA n-tiles per chunk
#define TBATCH  4                  // WMMA tiles per loop iteration
#define PFDIST  8                  // prefetch distance in 16-pt tiles

#define INF_F  (__builtin_inff())

// ---------------------------------------------------------------------------
// Kernel 1: shift = mean(q) - mean(tgt)   (3 floats into workspace)
// ---------------------------------------------------------------------------
__global__ void means_kernel(const float* __restrict__ q,
                             const float* __restrict__ tgt,
                             float* __restrict__ shift) {
    __shared__ float rx[256], ry[256], rz[256];
    const int t = threadIdx.x;
    float ax = 0.f, ay = 0.f, az = 0.f;
    for (int m = t; m < M_PTS; m += 256) {
        ax += q[3 * m] * (1.0f / M_PTS);
        ay += q[3 * m + 1] * (1.0f / M_PTS);
        az += q[3 * m + 2] * (1.0f / M_PTS);
    }
    for (int n = t; n < N_PTS; n += 256) {
        ax -= tgt[3 * n] * (1.0f / N_PTS);
        ay -= tgt[3 * n + 1] * (1.0f / N_PTS);
        az -= tgt[3 * n + 2] * (1.0f / N_PTS);
    }
    rx[t] = ax; ry[t] = ay; rz[t] = az;
    __syncthreads();
    for (int off = 128; off > 0; off >>= 1) {
        if (t < off) { rx[t] += rx[t + off]; ry[t] += ry[t + off]; rz[t] += rz[t + off]; }
        __syncthreads();
    }
    if (t == 0) { shift[0] = rx[0]; shift[1] = ry[0]; shift[2] = rz[0]; }
}

// ---------------------------------------------------------------------------
// Kernel 2: partial top-32 per (16-query tile, N-chunk) using WMMA f32 16x16x4
//   A (16x4) = -2 * [qx qy qz 0]    B (4x16) = [rx ry rz 0]^T
//   C        = |q|^2 + |r|^2        -> D = d2 block directly
// TBATCH=4 independent WMMAs per iteration (shared A, independent B/C/D ->
// no D->A/B RAW hazard, back-to-back matrix-pipe issue), then one barrier
// and a 64-candidate selection pass.
// ---------------------------------------------------------------------------
__global__ void topk_partial_kernel(const float* __restrict__ q,
                                    const float* __restrict__ tgt,
                                    const float* __restrict__ shift,
                                    float* __restrict__ pd2,
                                    int* __restrict__ pidx) {
    const int lane  = threadIdx.x;   // block = 1 wave (32)
    const int mtile = blockIdx.x;    // 0..255
    const int chunk = blockIdx.y;    // 0..7
    const int mm    = lane & 15;
    const int hi    = lane >> 4;

    __shared__ float sqx[16], sqy[16], sqz[16], sqn[16];
    __shared__ float sd[TBATCH * 256];       // 4x 16x16 d2 staging
    __shared__ float td[16 * NSAMPLE];       // reservoir distances
    __shared__ int   ti[16 * NSAMPLE];       // reservoir indices

    const float sx = shift[0], sy = shift[1], sz = shift[2];

    if (lane < 16) {
        const int m = mtile * 16 + lane;
        const float x = q[3 * m], y = q[3 * m + 1], z = q[3 * m + 2];
        sqx[lane] = x; sqy[lane] = y; sqz[lane] = z;
        sqn[lane] = x * x + y * y + z * z;
    }
    for (int i = lane; i < 16 * NSAMPLE; i += 32) { td[i] = INF_F; ti[i] = 0; }
    __syncthreads();

    v2f a;
    a.x = -2.0f * (hi ? sqz[mm] : sqx[mm]);
    a.y = hi ? 0.0f : (-2.0f * sqy[mm]);
    float cq[8];
#pragma unroll
    for (int v = 0; v < 8; ++v) cq[v] = sqn[v + 8 * hi];

    float curMax = INF_F;            // selection state (lanes 0..15 own query=lane)
    int   maxPos = 0;

    const int n_base = chunk * CHUNK_N;
    for (int it = 0; it < NTILES; it += TBATCH) {
        // prefetch a future tile's xyz line into near caches
        if (it + PFDIST < NTILES) {
            const int npf = n_base + (it + PFDIST) * 16 + mm;
            __builtin_prefetch(&tgt[3 * npf], 0, 1);
        }

        // load TBATCH * 16 ref points (both half-waves fetch the same 16 ->
        // coalesced, WGP$ hit for the second half)
        float px[TBATCH], py[TBATCH], pz[TBATCH], rn[TBATCH];
#pragma unroll
        for (int t = 0; t < TBATCH; ++t) {
            const int nl = n_base + (it + t) * 16 + mm;
            px[t] = tgt[3 * nl] + sx;
            py[t] = tgt[3 * nl + 1] + sy;
            pz[t] = tgt[3 * nl + 2] + sz;
            rn[t] = px[t] * px[t] + py[t] * py[t] + pz[t] * pz[t];
        }

        // 4 independent WMMAs: D = (|q|^2+|r|^2) - 2 q.r
#pragma unroll
        for (int t = 0; t < TBATCH; ++t) {
            v2f b;
            b.x = hi ? pz[t] : px[t];
            b.y = hi ? 0.0f : py[t];
            v8f c;
#pragma unroll
            for (int v = 0; v < 8; ++v) c[v] = cq[v] + rn[t];
            v8f d = __builtin_amdgcn_wmma_f32_16x16x4_f32(
                false, a, false, b, (short)0, c, false, false);
#pragma unroll
            for (int v = 0; v < 8; ++v)
                sd[t * 256 + (v + 8 * hi) * 16 + mm] = d[v];
        }
        __syncthreads();

        if (lane < 16) {             // lane owns one query row; serial reservoir insert
            for (int t = 0; t < TBATCH; ++t) {
                const int n0 = n_base + (it + t) * 16;
                for (int j = 0; j < 16; ++j) {
                    const float dd = sd[t * 256 + lane * 16 + j];
                    if (dd < curMax) {
                        td[lane * NSAMPLE + maxPos] = dd;
                        ti[lane * NSAMPLE + maxPos] = n0 + j;
                        float nm = -INF_F; int np = 0;
                        for (int p = 0; p < NSAMPLE; ++p) {
                            const float tv = td[lane * NSAMPLE + p];
                            if (tv > nm) { nm = tv; np = p; }
                        }
                        curMax = nm; maxPos = np;
                    }
                }
            }
        }
        __syncthreads();
    }

    if (lane < 16) {
        const int m = mtile * 16 + lane;
        const long base = ((long)m * NCHUNK + chunk) * NSAMPLE;
        for (int s = 0; s < NSAMPLE; ++s) {
            pd2[base + s]  = td[lane * NSAMPLE + s];
            pidx[base + s] = ti[lane * NSAMPLE + s];
        }
    }
}

// ---------------------------------------------------------------------------
// Kernel 3: merge 8 partial lists (256 candidates) -> final 32 per query,
//           write final indices + mean of gathered target points (output 1)
// ---------------------------------------------------------------------------
__global__ void merge_kernel(const float* __restrict__ pd2,
                             const int* __restrict__ pidx,
                             const float* __restrict__ tgt,
                             int* __restrict__ fidx,
                             float* __restrict__ out_pts) {
    const int m = blockIdx.x;
    const int lane = threadIdx.x;    // 32
    __shared__ float cd[NCHUNK * NSAMPLE];
    __shared__ int   ci[NCHUNK * NSAMPLE];
    __shared__ float r0[32], r1[32], r2[32];

    for (int i = lane; i < NCHUNK * NSAMPLE; i += 32) {
        cd[i] = pd2[(long)m * (NCHUNK * NSAMPLE) + i];
        ci[i] = pidx[(long)m * (NCHUNK * NSAMPLE) + i];
    }
    __syncthreads();

    float ax = 0.f, ay = 0.f, az = 0.f;
    for (int i = lane; i < NCHUNK * NSAMPLE; i += 32) {
        const float di = cd[i];
        int rank = 0;
        for (int j = 0; j < NCHUNK * NSAMPLE; ++j) {
            const float dj = cd[j];
            rank += (dj < di) || (dj == di && j < i);   // strict total order
        }
        if (rank < NSAMPLE) {                           // ranks are distinct
            const int n = ci[i];
            fidx[m * NSAMPLE + rank] = n;
            ax += tgt[3 * n]; ay += tgt[3 * n + 1]; az += tgt[3 * n + 2];
        }
    }
    r0[lane] = ax; r1[lane] = ay; r2[lane] = az;
    __syncthreads();
    for (int off = 16; off > 0; off >>= 1) {
        if (lane < off) { r0[lane] += r0[lane + off]; r1[lane] += r1[lane + off]; r2[lane] += r2[lane + off]; }
        __syncthreads();
    }
    if (lane == 0) {
        const float inv = 1.0f / NSAMPLE;
        out_pts[3 * m]     = r0[0] * inv;
        out_pts[3 * m + 1] = r1[0] * inv;
        out_pts[3 * m + 2] = r2[0] * inv;
    }
}

// ---------------------------------------------------------------------------
// Kernel 4: feature max-pool over the 32 neighbors (output 2).
// feat is (F, N); whole array (33.5 MB) is L2-resident on MI455X (192 MB L2),
// so the scattered gathers stay on-chip.
// ---------------------------------------------------------------------------
__global__ void featmax_kernel(const float* __restrict__ feat,
                               const int* __restrict__ fidx,
                               float* __restrict__ out_feat) {
    const int m = blockIdx.x;
    const int f = threadIdx.x;       // 256
    __shared__ int sidx[NSAMPLE];
    if (f < NSAMPLE) sidx[f] = fidx[m * NSAMPLE + f];
    __syncthreads();
    float r = -INF_F;
#pragma unroll 8
    for (int s = 0; s < NSAMPLE; ++s) {
        r = fmaxf(r, feat[(long)f * N_PTS + sidx[s]]);
    }
    out_feat[(long)m * F_DIM + f] = r;
}

// ---------------------------------------------------------------------------
extern "C" void kernel_launch(void* const* d_in, const int* in_sizes, int n_in,
                              void* d_out, int out_size, void* d_ws, size_t ws_size,
                              hipStream_t stream) {
    const float* q    = (const float*)d_in[0];  // candidate_pts (1,128,32,3)
    // d_in[1] (src_keypts) is unused by the reference computation
    const float* tgt  = (const float*)d_in[2];  // tgt_pts_xyz (1,32768,3)
    const float* feat = (const float*)d_in[3];  // tgt_deep_feat_pts (1,256,32768)
    float* out = (float*)d_out;                 // [pts 12288 | feats 1048576]

    char* ws = (char*)d_ws;
    const size_t PART = (size_t)M_PTS * NCHUNK * NSAMPLE;  // 1,048,576 entries
    float* shift = (float*)ws;                             // 16 B
    float* pd2   = (float*)(ws + 16);                      // 4 MB
    int*   pidx  = (int*)(ws + 16 + PART * 4);             // 4 MB
    int*   fidx  = (int*)(ws + 16 + 2 * PART * 4);         // 512 KB

    means_kernel<<<1, 256, 0, stream>>>(q, tgt, shift);
    topk_partial_kernel<<<dim3(M_PTS / 16, NCHUNK), 32, 0, stream>>>(q, tgt, shift, pd2, pidx);
    merge_kernel<<<M_PTS, 32, 0, stream>>>(pd2, pidx, tgt, fidx, out);
    featmax_kernel<<<M_PTS, F_DIM, 0, stream>>>(feat, fidx, out + (size_t)M_PTS * 3);
}